// CapsuleBase_6262062317710
// MI455X (gfx1250) — compile-verified
//
#include <hip/hip_runtime.h>
#include <hip/hip_bf16.h>

// ---------------- problem dimensions ----------------
static constexpr int  kN    = 40000;
static constexpr int  kInit = 100;
static constexpr int  kD    = 200;
static constexpr int  kF    = 4;
static constexpr int  kE    = 800000;
static constexpr int  kNrel = 800;
static constexpr int  kB    = 2048;
static constexpr int  kH2   = 100;
static constexpr int  kFD   = kF * kD;      // 800
static constexpr int  kMaxK = 200;          // largest K of any GEMM call

static constexpr int ACT_NONE = 0, ACT_TANH = 1, ACT_LEAKY = 2, ACT_RELU = 3;

typedef float    v2f   __attribute__((ext_vector_type(2)));
typedef float    v8f   __attribute__((ext_vector_type(8)));
typedef unsigned u32x4 __attribute__((ext_vector_type(4)));
typedef int      i32x4 __attribute__((ext_vector_type(4)));
typedef int      i32x8 __attribute__((ext_vector_type(8)));

#if defined(__AMDGCN__) &&                                   \
    __has_builtin(__builtin_amdgcn_tensor_load_to_lds) &&    \
    __has_builtin(__builtin_amdgcn_s_wait_tensorcnt)
#define USE_TDM 1
#else
#define USE_TDM 0
#endif

// ---------------- WMMA f32 16x16x4 (native fp32 matrix op on CDNA5) --------
// A 16x4 (ISA 7.12.2): lanes 0-15 row M=lane, V0=K0 V1=K1; lanes 16-31 K2/K3.
// B 4x16 mirrored (N=lane&15). C/D: VGPR v -> M=v (+8 upper half), N=lane&15.
__device__ __forceinline__ v8f wmma_f32_16x16x4(v2f a, v2f b, v8f c) {
#if defined(__AMDGCN__) && __has_builtin(__builtin_amdgcn_wmma_f32_16x16x4_f32)
  return __builtin_amdgcn_wmma_f32_16x16x4_f32(false, a, false, b,
                                               (short)0, c, false, false);
#else
  int lane = threadIdx.x & 31;
#pragma unroll
  for (int v = 0; v < 8; ++v) {
    int m = v + ((lane & 16) >> 1);
    int n = lane & 15;
    float acc = c[v];
#pragma unroll
    for (int k = 0; k < 4; ++k) {
      float av = __shfl((k & 1) ? a.y : a.x, (m & 15) + ((k & 2) ? 16 : 0), 32);
      float bv = __shfl((k & 1) ? b.y : b.x, n + ((k & 2) ? 16 : 0), 32);
      acc = fmaf(av, bv, acc);
    }
    c[v] = acc;
  }
  return c;
#endif
}

// ---------------- generic WMMA GEMM: C = act(A @ B (+ bias)) ---------------
// A: MxK row-major (lda), B: KxN row-major (ldb), C: MxN row-major (ldc).
// M % 16 == 0 and K % 4 == 0 at every call site; ragged N edge handled by
// zero-filled LDS staging (loads) + predicated stores.
// Block = 256 threads = 8 waves: one 16-col B panel staged in LDS (via the
// Tensor Data Mover when available), 8 M-tiles consume it.
template <int ACT, int BIAS>
__global__ void gemm16_kernel(const float* __restrict__ A, int lda,
                              const float* __restrict__ Bm, int ldb,
                              const float* __restrict__ bias,
                              float* __restrict__ C, int ldc,
                              int M, int Nc, int K) {
  __shared__ float ldsB[kMaxK * 16];

  const int tilesN = (Nc + 15) >> 4;
  const int tilesM = M >> 4;
  const int tn  = blockIdx.x % tilesN;
  const int tmg = blockIdx.x / tilesN;          // group of 8 M-tiles
  const int n0  = tn << 4;
  const int lane = threadIdx.x & 31;

  // ---- stage B[K x 16] panel (columns n0..n0+15, OOB columns = 0) ----
#if USE_TDM
  if ((threadIdx.x >> 5) == 0) {                 // one wave issues the DMA
    const unsigned lds_off = (unsigned)(uintptr_t)ldsB;   // low 32b = LDS addr
    const unsigned long long ga = (unsigned long long)(uintptr_t)(Bm + n0);
    const unsigned rem = (unsigned)(Nc - n0);    // remaining cols from tile start
    // D# group0: count=1 | lds_addr | global_addr[56:0] | type=2 (ISA 8.3)
    u32x4 g0;
    g0[0] = 1u;
    g0[1] = lds_off;
    g0[2] = (unsigned)(ga & 0xFFFFFFFFu);
    g0[3] = (unsigned)((ga >> 32) & 0x01FFFFFFu) | (2u << 30);
    // D# group1 (ISA 8.4): data_size=2 (4B); tensor_dim0=rem, tensor_dim1=K,
    // tile_dim0=16, tile_dim1=K, tile_dim2=0, dim0_stride=ldb, dim1_stride=0
    i32x8 g1;
    g1[0] = 0x00020000;                          // data_size=2, no pad/iterate
    g1[1] = (int)((rem & 0xFFFFu) << 16);        // tensor_dim0[15:0] @ bits63:48
    g1[2] = (int)(((rem >> 16) & 0xFFFFu) |      // tensor_dim0[31:16]
                  (((unsigned)K & 0xFFFFu) << 16)); // tensor_dim1[15:0]
    g1[3] = (int)((((unsigned)K >> 16) & 0xFFFFu) |
                  (16u << 16));                  // tile_dim0 = 16
    g1[4] = (int)((unsigned)K & 0xFFFFu);        // tile_dim1 = K, tile_dim2 = 0
    g1[5] = ldb;                                 // dim0 stride low 32 (elements)
    g1[6] = 0;                                   // dim0 stride hi / dim1 stride lo
    g1[7] = 0;
    const i32x4 z4 = {0, 0, 0, 0};
#if __clang_major__ >= 23
    const i32x8 z8 = {0, 0, 0, 0, 0, 0, 0, 0};
    __builtin_amdgcn_tensor_load_to_lds(g0, g1, z4, z4, z8, 0);
#else
    __builtin_amdgcn_tensor_load_to_lds(g0, g1, z4, z4, 0);
#endif
    __builtin_amdgcn_s_wait_tensorcnt(0);
  }
#else
  for (int t = threadIdx.x; t < K * 16; t += 256) {
    const int k = t >> 4, c = t & 15;
    ldsB[t] = (n0 + c < Nc) ? Bm[(size_t)k * ldb + n0 + c] : 0.0f;
  }
#endif
  __syncthreads();

  const int tm = tmg * 8 + (threadIdx.x >> 5);
  if (tm >= tilesM) return;
  const int m0 = tm << 4;
  const int row = m0 + (lane & 15);
  const int nlane = lane & 15;
  const int khalf = (lane >> 4) << 1;
  const float* __restrict__ arow = A + (size_t)row * lda;

  v8f c = {};
  for (int k0 = 0; k0 < K; k0 += 4) {
    v2f a, b;
    a.x = arow[k0 + khalf];
    a.y = arow[k0 + khalf + 1];
    b.x = ldsB[(k0 + khalf) * 16 + nlane];
    b.y = ldsB[(k0 + khalf + 1) * 16 + nlane];
    c = wmma_f32_16x16x4(a, b, c);
  }

  const int col = n0 + nlane;
  if (col >= Nc) return;
  const float bv = BIAS ? bias[col] : 0.0f;
  const int mrow0 = m0 + ((lane >> 4) << 3);
#pragma unroll
  for (int v = 0; v < 8; ++v) {
    float val = c[v] + bv;
    if (ACT == ACT_TANH)       val = tanhf(val);
    else if (ACT == ACT_LEAKY) val = (val > 0.0f) ? val : 0.2f * val;
    else if (ACT == ACT_RELU)  val = fmaxf(val, 0.0f);
    C[(size_t)(mrow0 + v) * ldc + col] = val;
  }
}

// ---------------- small utility kernels ----------------
__global__ void fill_u32_kernel(unsigned* __restrict__ p, unsigned v, long n) {
  long t = (long)blockIdx.x * blockDim.x + threadIdx.x;
  if (t < n) p[t] = v;
}

// softmax over the F=4 capsule axis: kw[n, f*D+d], softmax across f.
__global__ void softmax_f_kernel(float* __restrict__ kw) {
  long t = (long)blockIdx.x * blockDim.x + threadIdx.x;
  if (t >= (long)kN * kD) return;
  const long n = t / kD;
  const int d  = (int)(t % kD);
  float* p = kw + n * (size_t)kFD + d;
  float v0 = p[0], v1 = p[kD], v2 = p[2 * kD], v3 = p[3 * kD];
  float mx = fmaxf(fmaxf(v0, v1), fmaxf(v2, v3));
  float e0 = __expf(v0 - mx), e1 = __expf(v1 - mx);
  float e2 = __expf(v2 - mx), e3 = __expf(v3 - mx);
  float inv = 1.0f / (e0 + e1 + e2 + e3);
  p[0] = e0 * inv; p[kD] = e1 * inv; p[2 * kD] = e2 * inv; p[3 * kD] = e3 * inv;
}

// ordered-uint encoding for float atomicMax
__device__ __forceinline__ unsigned ford(float f) {
  unsigned u = __float_as_uint(f);
  return (u & 0x80000000u) ? ~u : (u | 0x80000000u);
}
__device__ __forceinline__ float funord(unsigned u) {
  u = (u & 0x80000000u) ? (u & 0x7FFFFFFFu) : ~u;
  return __uint_as_float(u);
}

// wave-per-edge: logit[e,k] = leaky(sum_d x[src,k,d]*r[et,d]*x[dst,k,d]);
// segment-max over dst via ordered-uint atomicMax.
__global__ void edge_logit_kernel(const int* __restrict__ eidx,
                                  const int* __restrict__ etype,
                                  const float* __restrict__ x,
                                  const float* __restrict__ r,
                                  float* __restrict__ logits,
                                  unsigned* __restrict__ segmax) {
  const long e = ((long)blockIdx.x * blockDim.x + threadIdx.x) >> 5;
  const int lane = threadIdx.x & 31;
  if (e >= kE) return;
  const int src = eidx[e];
  const int dst = eidx[kE + e];
  const float* __restrict__ re = r + (size_t)etype[e] * kD;
  const float* __restrict__ xs = x + (size_t)src * kFD;
  const float* __restrict__ xd = x + (size_t)dst * kFD;
  __builtin_prefetch(re, 0, 0);
#pragma unroll
  for (int k = 0; k < kF; ++k) {
    float part = 0.0f;
    for (int d = lane; d < kD; d += 32)
      part += xs[k * kD + d] * re[d] * xd[k * kD + d];
    for (int off = 16; off; off >>= 1) part += __shfl_down(part, off, 32);
    if (lane == 0) {
      float lg = (part > 0.0f) ? part : 0.2f * part;
      logits[e * kF + k] = lg;
      atomicMax(segmax + (size_t)dst * kF + k, ford(lg));
    }
  }
}

// thread per (edge,k): a = exp(logit - segmax[dst,k]); den[dst,k] += a
__global__ void edge_den_kernel(const int* __restrict__ eidx,
                                float* __restrict__ logits,
                                const unsigned* __restrict__ segmax,
                                float* __restrict__ den) {
  const long t = (long)blockIdx.x * blockDim.x + threadIdx.x;
  if (t >= (long)kE * kF) return;
  const long e = t >> 2;
  const int k = (int)(t & 3);
  const int dst = eidx[kE + e];
  const float mx = funord(segmax[(size_t)dst * kF + k]);
  const float a = __expf(logits[t] - mx);
  logits[t] = a;                       // reuse buffer for the numerator
  atomicAdd(den + (size_t)dst * kF + k, a);
}

// wave-per-edge: agg[dst,k,d] += (a/(den+eps)) * x[src,k,d]*r[et,d]
__global__ void edge_scatter_kernel(const int* __restrict__ eidx,
                                    const int* __restrict__ etype,
                                    const float* __restrict__ x,
                                    const float* __restrict__ r,
                                    const float* __restrict__ logits,
                                    const float* __restrict__ den,
                                    float* __restrict__ agg) {
  const long e = ((long)blockIdx.x * blockDim.x + threadIdx.x) >> 5;
  const int lane = threadIdx.x & 31;
  if (e >= kE) return;
  const int src = eidx[e];
  const int dst = eidx[kE + e];
  const float* __restrict__ re = r + (size_t)etype[e] * kD;
  const float* __restrict__ xs = x + (size_t)src * kFD;
  float* __restrict__ ad = agg + (size_t)dst * kFD;
#pragma unroll
  for (int k = 0; k < kF; ++k) {
    const float coeff =
        logits[e * kF + k] / (den[(size_t)dst * kF + k] + 1e-16f);
    for (int d = lane; d < kD; d += 32)
      atomicAdd(&ad[k * kD + d], coeff * xs[k * kD + d] * re[d]);
  }
}

// x = tanh(agg) * kw ; also reset agg for the next layer/call
__global__ void update_x_kernel(float* __restrict__ x,
                                float* __restrict__ agg,
                                const float* __restrict__ kw) {
  const long t = (long)blockIdx.x * blockDim.x + threadIdx.x;
  if (t >= (long)kN * kFD) return;
  x[t] = tanhf(agg[t]) * kw[t];
  agg[t] = 0.0f;
}

// sub_emb[b,fd] = x[sub[b],fd]; rel_emb[b,fd] = init_rel[rel[b], fd % D]
__global__ void gather_kernel(const int* __restrict__ sub,
                              const int* __restrict__ rel,
                              const float* __restrict__ x,
                              const float* __restrict__ init_rel,
                              float* __restrict__ sub_emb,
                              float* __restrict__ rel_emb) {
  const long t = (long)blockIdx.x * blockDim.x + threadIdx.x;
  if (t >= (long)kB * kFD) return;
  const int b = (int)(t / kFD);
  const int fd = (int)(t % kFD);
  sub_emb[t] = x[(size_t)sub[b] * kFD + fd];
  rel_emb[t] = init_rel[(size_t)rel[b] * kD + (fd % kD)];
}

// mi += sum_{b,d} ((mu-yperm)^2 - (mu-y)^2) * exp(-logvar) * 0.5/B
__global__ void club_loss_kernel(const float* __restrict__ mu,
                                 const float* __restrict__ lv,
                                 const float* __restrict__ sub_emb,
                                 const int* __restrict__ perm,
                                 int joff, float* __restrict__ mi) {
  const long t = (long)blockIdx.x * blockDim.x + threadIdx.x;
  float contrib = 0.0f;
  if (t < (long)kB * kD) {
    const int b = (int)(t / kD);
    const int d = (int)(t % kD);
    const float m = mu[t];
    const float iv = __expf(-lv[t]);
    const float y  = sub_emb[(size_t)b * kFD + joff + d];
    const float yp = sub_emb[(size_t)perm[b] * kFD + joff + d];
    const float dp = m - y, dn = m - yp;
    contrib = (dn * dn - dp * dp) * iv;
  }
  for (int off = 16; off; off >>= 1) contrib += __shfl_down(contrib, off, 32);
  if ((threadIdx.x & 31) == 0)
    atomicAdd(mi, contrib * (0.5f / (float)kB));
}

// ---------------- host-side dispatch ----------------
static inline void launch_gemm(int act, int hasb, const float* A, int lda,
                               const float* Bm, int ldb, const float* bias,
                               float* C, int ldc, int M, int Nc, int K,
                               hipStream_t s) {
  const int tilesN = (Nc + 15) / 16;
  const int tmGroups = (M / 16 + 7) / 8;
  dim3 g(tmGroups * tilesN), b(256);
#define GEMM_CASE(A_, B_) \
  gemm16_kernel<A_, B_><<<g, b, 0, s>>>(A, lda, Bm, ldb, bias, C, ldc, M, Nc, K)
  if (hasb) {
    switch (act) {
      case ACT_TANH: GEMM_CASE(ACT_TANH, 1); break;
      case ACT_RELU: GEMM_CASE(ACT_RELU, 1); break;
      default:       GEMM_CASE(ACT_NONE, 1); break;
    }
  } else {
    switch (act) {
      case ACT_LEAKY: GEMM_CASE(ACT_LEAKY, 0); break;
      default:        GEMM_CASE(ACT_NONE, 0); break;
    }
  }
#undef GEMM_CASE
}

static inline void launch_fill(void* p, unsigned v, long n, hipStream_t s) {
  const long blocks = (n + 255) / 256;
  fill_u32_kernel<<<(int)blocks, 256, 0, s>>>((unsigned*)p, v, n);
}

extern "C" void kernel_launch(void* const* d_in, const int* in_sizes, int n_in,
                              void* d_out, int out_size, void* d_ws,
                              size_t ws_size, hipStream_t stream) {
  (void)in_sizes; (void)n_in; (void)out_size; (void)ws_size;
  const int*   sub        = (const int*)d_in[0];
  const int*   rel        = (const int*)d_in[1];
  const int*   eidx       = (const int*)d_in[2];   // (2,E)
  const int*   etype      = (const int*)d_in[3];
  const int*   perm       = (const int*)d_in[4];
  const float* init_embed = (const float*)d_in[5];
  const float* init_rel   = (const float*)d_in[6];
  const float* pca_w      = (const float*)d_in[7];
  const float* pca_b      = (const float*)d_in[8];
  const float* cw_w       = (const float*)d_in[9];
  const float* w_rel      = (const float*)d_in[10];
  const float* mu_w1      = (const float*)d_in[11];
  const float* mu_b1      = (const float*)d_in[12];
  const float* mu_w2      = (const float*)d_in[13];
  const float* mu_b2      = (const float*)d_in[14];
  const float* lv_w1      = (const float*)d_in[15];
  const float* lv_b1      = (const float*)d_in[16];
  const float* lv_w2      = (const float*)d_in[17];
  const float* lv_b2      = (const float*)d_in[18];

  // outputs: sub_emb (B,FD), rel_emb (B,FD), x (N,FD), mi_loss (1)
  float* out     = (float*)d_out;
  float* sub_emb = out;
  float* rel_emb = out + (size_t)kB * kFD;
  float* x       = out + 2 * (size_t)kB * kFD;
  float* mi      = x + (size_t)kN * kFD;

  // workspace layout (floats)
  float*    ws     = (float*)d_ws;
  float*    kw     = ws;                           // 32,000,000
  float*    agg    = ws + 32000000L;               // 32,000,000
  float*    r1     = ws + 64000000L;               // 160,000
  float*    r2     = ws + 64160000L;               // 160,000
  float*    logits = ws + 64320000L;               // 3,200,000  (E*F)
  unsigned* segmax = (unsigned*)(ws + 67520000L);  // 160,000 (N*F)
  float*    den    = ws + 67680000L;               // 160,000
  float*    hbuf   = ws + 67840000L;               // 204,800  (B*H2)
  float*    mubuf  = ws + 68044800L;               // 409,600  (B*D)
  float*    lvbuf  = ws + 68454400L;               // 409,600

  // ---- encode ----
  // x = tanh(init_embed @ pca_w + pca_b)           (40000 x 800)
  launch_gemm(ACT_TANH, 1, init_embed, kInit, pca_w, kFD, pca_b,
              x, kFD, kN, kFD, kInit, stream);
  // kw = leaky(x.reshape(-1,D) @ cw_w)             (160000 x 200)
  launch_gemm(ACT_LEAKY, 0, x, kD, cw_w, kD, nullptr,
              kw, kD, kN * kF, kD, kD, stream);
  // kw = softmax over F
  softmax_f_kernel<<<(int)(((long)kN * kD + 255) / 256), 256, 0, stream>>>(kw);

  launch_fill(agg, 0u, (long)kN * kFD, stream);

  const int edgeBlocks = (int)(((long)kE * 32 + 255) / 256);
  for (int l = 0; l < 2; ++l) {
    const float* rcur = (l == 0) ? init_rel : r1;
    float* rnext = (l == 0) ? r1 : r2;
    launch_fill(segmax, 0u, (long)kN * kF, stream);      // minimal ordered enc
    launch_fill(den, 0u, (long)kN * kF, stream);
    edge_logit_kernel<<<edgeBlocks, 256, 0, stream>>>(eidx, etype, x, rcur,
                                                      logits, segmax);
    edge_den_kernel<<<(int)(((long)kE * kF + 255) / 256), 256, 0, stream>>>(
        eidx, logits, segmax, den);
    edge_scatter_kernel<<<edgeBlocks, 256, 0, stream>>>(eidx, etype, x, rcur,
                                                        logits, den, agg);
    update_x_kernel<<<(int)(((long)kN * kFD + 255) / 256), 256, 0, stream>>>(
        x, agg, kw);
    // r = r @ w_rel[l]
    launch_gemm(ACT_NONE, 0, rcur, kD, w_rel + (size_t)l * kD * kD, kD,
                nullptr, rnext, kD, kNrel, kD, kD, stream);
  }

  // ---- gathers + loss ----
  gather_kernel<<<(int)(((long)kB * kFD + 255) / 256), 256, 0, stream>>>(
      sub, rel, x, init_rel, sub_emb, rel_emb);
  launch_fill(mi, 0u, 1, stream);

  int cnt = 0;
  for (int i = 0; i < kF; ++i) {
    for (int j = i + 1; j < kF; ++j, ++cnt) {
      const float* xi = sub_emb + (size_t)i * kD;   // lda = kFD
      // mu = relu(xi@w1+b1)@w2 + b2
      launch_gemm(ACT_RELU, 1, xi, kFD, mu_w1 + (size_t)cnt * kD * kH2, kH2,
                  mu_b1 + (size_t)cnt * kH2, hbuf, kH2, kB, kH2, kD, stream);
      launch_gemm(ACT_NONE, 1, hbuf, kH2, mu_w2 + (size_t)cnt * kH2 * kD, kD,
                  mu_b2 + (size_t)cnt * kD, mubuf, kD, kB, kD, kH2, stream);
      // logvar = tanh(relu(xi@v1+c1)@v2 + c2)
      launch_gemm(ACT_RELU, 1, xi, kFD, lv_w1 + (size_t)cnt * kD * kH2, kH2,
                  lv_b1 + (size_t)cnt * kH2, hbuf, kH2, kB, kH2, kD, stream);
      launch_gemm(ACT_TANH, 1, hbuf, kH2, lv_w2 + (size_t)cnt * kH2 * kD, kD,
                  lv_b2 + (size_t)cnt * kD, lvbuf, kD, kB, kD, kH2, stream);
      club_loss_kernel<<<(int)(((long)kB * kD + 255) / 256), 256, 0, stream>>>(
          mubuf, lvbuf, sub_emb, perm, j * kD, mi);
    }
  }
}